// URLGNN_68436008894533
// MI455X (gfx1250) — compile-verified
//
#include <hip/hip_runtime.h>

#define N_NODES  100000
#define N_EDGES  640000
#define N_GRAPHS 2000
#define VOCAB    128
#define EMBED    64
#define H1       128
#define H2       64

typedef __attribute__((ext_vector_type(2))) float v2f;
typedef __attribute__((ext_vector_type(8))) float v8f;

// ---------------- generic fill ----------------
__global__ void gnn_fill(float* __restrict__ p, float v, int n) {
    int i = blockIdx.x * blockDim.x + threadIdx.x;
    int stride = gridDim.x * blockDim.x;
    for (; i < n; i += stride) p[i] = v;
}

// ---------------- T1 = emb @ W1 (128x128 table, folds the layer-1 GEMM) ----
__global__ void gnn_table(const float* __restrict__ emb,
                          const float* __restrict__ W1,
                          float* __restrict__ T1) {
    int r = blockIdx.x;    // vocab row
    int c = threadIdx.x;   // H1 column
    float acc = 0.f;
#pragma unroll 8
    for (int k = 0; k < EMBED; ++k)
        acc += emb[r * EMBED + k] * W1[k * H1 + c];
    T1[r * H1 + c] = acc;
}

// ---------------- degree (in-edges on dst) ---------------------------------
__global__ void gnn_deg(const int* __restrict__ dst, float* __restrict__ deg, int ne) {
    int i = blockIdx.x * blockDim.x + threadIdx.x;
    int stride = gridDim.x * blockDim.x;
    for (; i < ne; i += stride) atomicAdd(&deg[dst[i]], 1.0f);
}

__global__ void gnn_rsqrt(const float* __restrict__ deg, float* __restrict__ dinv, int n) {
    int i = blockIdx.x * blockDim.x + threadIdx.x;
    int stride = gridDim.x * blockDim.x;
    for (; i < n; i += stride) dinv[i] = rsqrtf(deg[i]);
}

// ---------------- layer-1 edge scatter: one wave32 per edge, 4 feats/lane --
__global__ void gnn_scatter1(const int* __restrict__ src, const int* __restrict__ dst,
                             const int* __restrict__ x, const float* __restrict__ T1,
                             const float* __restrict__ dinv, float* __restrict__ agg1) {
    int gtid = blockIdx.x * blockDim.x + threadIdx.x;
    int e    = gtid >> 5;
    int lane = gtid & 31;
    if (e >= N_EDGES) return;
    int s = src[e], d = dst[e];
    float nrm = dinv[s] * dinv[d];
    int t = x[s];
    float4 v = *(const float4*)(T1 + t * H1 + lane * 4);
    float* o = agg1 + (size_t)d * H1 + lane * 4;
    atomicAdd(o + 0, v.x * nrm);
    atomicAdd(o + 1, v.y * nrm);
    atomicAdd(o + 2, v.z * nrm);
    atomicAdd(o + 3, v.w * nrm);
}

// ---------------- layer-1 node pass: self term + bias + relu (in place) ----
__global__ void gnn_node1(const int* __restrict__ x, const float* __restrict__ T1,
                          const float* __restrict__ dinv, const float* __restrict__ b1,
                          float* __restrict__ agg1) {
    int i = blockIdx.x * blockDim.x + threadIdx.x;
    int stride = gridDim.x * blockDim.x;
    const int total = N_NODES * H1;
    for (; i < total; i += stride) {
        int n = i >> 7;          // / H1
        int j = i & (H1 - 1);    // % H1
        float di = dinv[n];
        float v = agg1[i] + T1[x[n] * H1 + j] * di * di + b1[j];
        agg1[i] = fmaxf(v, 0.0f);
    }
}

// ---------------- layer-2 GEMM via V_WMMA_F32_16X16X4_F32 ------------------
// Each wave owns a 16-row block of h1 and produces 16x64 of hw2.
// W2 is staged in LDS pair-interleaved: w2p[k/2][c] = {W2[k][c], W2[k+1][c]}
// so every WMMA B operand is one aligned ds_load_b64 into a VGPR pair.
__global__ void __launch_bounds__(128) gnn_gemm2(const float* __restrict__ h1,
                                                 const float* __restrict__ W2,
                                                 float* __restrict__ hw2) {
    __shared__ v2f w2p[(H1 / 2) * H2];   // 64x64 pairs = 32KB
    for (int i = threadIdx.x; i < (H1 / 2) * H2; i += 128) {
        int p = i >> 6;        // pair index = k/2
        int c = i & (H2 - 1);  // column
        w2p[i] = v2f{ W2[(2 * p) * H2 + c], W2[(2 * p + 1) * H2 + c] };
    }
    __syncthreads();

    const int waveid = threadIdx.x >> 5;
    const int lane   = threadIdx.x & 31;
    const int rb     = (blockIdx.x * 4 + waveid) * 16;  // row block base
    if (rb >= N_NODES) return;

    const int m      = lane & 15;        // M (or N) index within tile
    const int half   = lane >> 4;        // 0: K=0,1  1: K=2,3
    const int koff   = half << 1;        // 0 or 2

    v8f acc0 = {}, acc1 = {}, acc2 = {}, acc3 = {};
    const float* arow = h1 + (size_t)(rb + m) * H1 + koff;

#pragma unroll 4
    for (int k0 = 0; k0 < H1; k0 += 4) {
        v2f a = *(const v2f*)(arow + k0);          // A[m][k0+koff : +1]
        const v2f* brow = w2p + ((k0 >> 1) + half) * H2;
        v2f b0 = brow[ 0 + m];
        v2f b1v = brow[16 + m];
        v2f b2v = brow[32 + m];
        v2f b3v = brow[48 + m];
        acc0 = __builtin_amdgcn_wmma_f32_16x16x4_f32(false, a, false, b0,  (short)0, acc0, false, false);
        acc1 = __builtin_amdgcn_wmma_f32_16x16x4_f32(false, a, false, b1v, (short)0, acc1, false, false);
        acc2 = __builtin_amdgcn_wmma_f32_16x16x4_f32(false, a, false, b2v, (short)0, acc2, false, false);
        acc3 = __builtin_amdgcn_wmma_f32_16x16x4_f32(false, a, false, b3v, (short)0, acc3, false, false);
    }

    // D layout: VGPR i -> M = i (lanes 0-15) / 8+i (lanes 16-31), N = lane&15
    float* out = hw2 + (size_t)rb * H2;
    const int rofs = half << 3;          // 0 or 8
#pragma unroll
    for (int i = 0; i < 8; ++i) {
        int row = i + rofs;
        out[row * H2 +  0 + m] = acc0[i];
        out[row * H2 + 16 + m] = acc1[i];
        out[row * H2 + 32 + m] = acc2[i];
        out[row * H2 + 48 + m] = acc3[i];
    }
}

// ---------------- layer-2 edge scatter: one wave32 per edge, 2 feats/lane --
__global__ void gnn_scatter2(const int* __restrict__ src, const int* __restrict__ dst,
                             const float* __restrict__ hw2, const float* __restrict__ dinv,
                             float* __restrict__ agg2) {
    int gtid = blockIdx.x * blockDim.x + threadIdx.x;
    int e    = gtid >> 5;
    int lane = gtid & 31;
    if (e >= N_EDGES) return;
    int s = src[e], d = dst[e];
    float nrm = dinv[s] * dinv[d];
    float2 v = *(const float2*)(hw2 + (size_t)s * H2 + lane * 2);
    float* o = agg2 + (size_t)d * H2 + lane * 2;
    atomicAdd(o + 0, v.x * nrm);
    atomicAdd(o + 1, v.y * nrm);
}

// ---------------- layer-2 node pass + graph pooling ------------------------
__global__ void gnn_node2_pool(const float* __restrict__ hw2, const float* __restrict__ agg2,
                               const float* __restrict__ dinv, const float* __restrict__ b2,
                               const int* __restrict__ batch, float* __restrict__ outsum,
                               float* __restrict__ cnt) {
    int i = blockIdx.x * blockDim.x + threadIdx.x;
    int stride = gridDim.x * blockDim.x;
    const int total = N_NODES * H2;
    for (; i < total; i += stride) {
        int n = i >> 6;          // / H2
        int j = i & (H2 - 1);    // % H2
        float di = dinv[n];
        float v = agg2[i] + hw2[i] * di * di + b2[j];
        v = fmaxf(v, 0.0f);
        int g = batch[n];
        atomicAdd(&outsum[g * H2 + j], v);
        if (j == 0) atomicAdd(&cnt[g], 1.0f);
    }
}

__global__ void gnn_finalize(float* __restrict__ out, const float* __restrict__ cnt) {
    int i = blockIdx.x * blockDim.x + threadIdx.x;
    if (i >= N_GRAPHS * H2) return;
    out[i] /= fmaxf(cnt[i >> 6], 1.0f);
}

// ---------------------------------------------------------------------------
extern "C" void kernel_launch(void* const* d_in, const int* in_sizes, int n_in,
                              void* d_out, int out_size, void* d_ws, size_t ws_size,
                              hipStream_t stream) {
    const int*   x    = (const int*)d_in[0];
    const int*   ei   = (const int*)d_in[1];
    const int*   bat  = (const int*)d_in[2];
    const float* emb  = (const float*)d_in[3];
    const float* W1   = (const float*)d_in[4];
    const float* b1   = (const float*)d_in[5];
    const float* W2   = (const float*)d_in[6];
    const float* b2   = (const float*)d_in[7];
    float* out = (float*)d_out;

    const int* src = ei;
    const int* dst = ei + N_EDGES;

    // workspace layout (floats)
    float* ws   = (float*)d_ws;
    float* T1   = ws;                                 // 16384
    float* deg  = T1   + VOCAB * H1;                  // 100000
    float* dinv = deg  + N_NODES;                     // 100000
    float* agg1 = dinv + N_NODES;                     // N*H1 = 12.8M (becomes h1 in place)
    float* hw2  = agg1 + (size_t)N_NODES * H1;        // N*H2 = 6.4M
    float* agg2 = hw2  + (size_t)N_NODES * H2;        // N*H2 = 6.4M
    float* cnt  = agg2 + (size_t)N_NODES * H2;        // 2000

    // init
    gnn_fill<<<(N_NODES + 255) / 256, 256, 0, stream>>>(deg, 1.0f, N_NODES);
    gnn_fill<<<(N_NODES * H1 + 255) / 256, 256, 0, stream>>>(agg1, 0.0f, N_NODES * H1);
    gnn_fill<<<(N_NODES * H2 + 255) / 256, 256, 0, stream>>>(agg2, 0.0f, N_NODES * H2);
    gnn_fill<<<(N_GRAPHS + 255) / 256, 256, 0, stream>>>(cnt, 0.0f, N_GRAPHS);
    gnn_fill<<<(N_GRAPHS * H2 + 255) / 256, 256, 0, stream>>>(out, 0.0f, N_GRAPHS * H2);

    // T1 = emb @ W1 (collapses layer-1 GEMM into a 128x128 table)
    gnn_table<<<VOCAB, H1, 0, stream>>>(emb, W1, T1);

    // degrees + rsqrt
    gnn_deg<<<(N_EDGES + 255) / 256, 256, 0, stream>>>(dst, deg, N_EDGES);
    gnn_rsqrt<<<(N_NODES + 255) / 256, 256, 0, stream>>>(deg, dinv, N_NODES);

    // layer 1: scatter + node pass (relu, in place -> h1)
    gnn_scatter1<<<(N_EDGES * 32) / 256, 256, 0, stream>>>(src, dst, x, T1, dinv, agg1);
    gnn_node1<<<(N_NODES * H1 + 255) / 256, 256, 0, stream>>>(x, T1, dinv, b1, agg1);

    // layer 2 GEMM: hw2 = h1 @ W2 via WMMA f32 16x16x4
    int rowblocks = N_NODES / 16;                 // 6250
    gnn_gemm2<<<(rowblocks + 3) / 4, 128, 0, stream>>>(agg1, W2, hw2);

    // layer 2: scatter + node pass + pooling
    gnn_scatter2<<<(N_EDGES * 32) / 256, 256, 0, stream>>>(src, dst, hw2, dinv, agg2);
    gnn_node2_pool<<<(N_NODES * H2 + 255) / 256, 256, 0, stream>>>(hw2, agg2, dinv, b2, bat, out, cnt);
    gnn_finalize<<<(N_GRAPHS * H2 + 255) / 256, 256, 0, stream>>>(out, cnt);
}